// CustomAttention_45689862094989
// MI455X (gfx1250) — compile-verified
//
#include <hip/hip_runtime.h>

// ---------------------------------------------------------------------------
// CustomAttention for MI455X (gfx1250), fp32 end-to-end using
// V_WMMA_F32_16X16X4_F32 + GLOBAL_LOAD_ASYNC_TO_LDS_* (ASYNCcnt path),
// with double-buffered LDS so async copies overlap WMMA compute:
//   pipeline per tile:  s_wait_asynccnt 0 ; barrier ; issue(next) ; compute(cur)
//   (the wait drains loads that had a full compute phase to finish; one
//   barrier both publishes the arrived tile and licenses reuse of the other
//   buffer, since each wave signals only after its previous compute.)
//   Kernel 1: fused Q/K projection GEMM (x @ W^T + b), 64x128 block tiles,
//             2x2x{Q,K} register blocking per wave (8 WMMAs per fragment set).
//   Kernel 2: flash-attention (online softmax), 16 query rows / block,
//             key chunks of 128; first V slab prefetched during softmax.
// Workspace: Q (32MB) + K (32MB) fp32.
// ---------------------------------------------------------------------------

typedef float v2f __attribute__((ext_vector_type(2)));
typedef float v8f __attribute__((ext_vector_type(8)));

#define BATCH 2
#define SEQ   4096
#define DIM   1024
#define BSM   (BATCH * SEQ)   // 8192 rows for the projection GEMM

// LDS row pitch (floats): 80B rows -> 16B-aligned B128 async stores at col%4==0,
// and r*20 mod 64 banks is a full 16-cycle (ord(20 mod 64)=16) -> conflict-free.
#define LDP 20
#define SPITCH 132            // score/P row pitch (132 mod 64 = 4 -> conflict-free)

__device__ __forceinline__ v8f wmma_f32(v2f a, v2f b, v8f c) {
  // D = A(16x4) * B(4x16) + C(16x16), all fp32.
  return __builtin_amdgcn_wmma_f32_16x16x4_f32(
      false, a, false, b, (short)0, c, false, false);
}

// Async global->LDS copies (VGLOBAL async encoding, tracked by ASYNCcnt).
// the LDS dest address comes from a per-lane VGPR, so pass the low 32 bits of the generic ptr.
__device__ __forceinline__ void async_ld_b128(const float* lds_dst, const float* gsrc) {
  asm volatile("global_load_async_to_lds_b128 %0, %1, off"
               :: "v"((unsigned)(unsigned long long)lds_dst),
                  "v"((unsigned long long)gsrc)
               : "memory");
}
__device__ __forceinline__ void async_ld_b32(const float* lds_dst, const float* gsrc) {
  asm volatile("global_load_async_to_lds_b32 %0, %1, off"
               :: "v"((unsigned)(unsigned long long)lds_dst),
                  "v"((unsigned long long)gsrc)
               : "memory");
}
__device__ __forceinline__ void wait_async0() {
  asm volatile("s_wait_asynccnt 0x0" ::: "memory");
}

// ---------------------------------------------------------------------------
// Kernel 1: Q = X @ Wq^T + bq ; K = X @ Wk^T + bk
// C[m][n] = sum_d X[m][d] * W[n][d]; block tile 64(M) x 128(N), K-step 16.
// ---------------------------------------------------------------------------
__device__ __forceinline__ void k1_issue(
    const float* X, const float* Wq, const float* Wk,
    float* xb, float* wqb, float* wkb,
    int m0b, int n0b, int k0, int tid) {
  const int xr = tid >> 2, xc = (tid & 3) * 4;   // 256 float4 slots
  async_ld_b128(&xb[xr * LDP + xc], &X[(size_t)(m0b + xr) * DIM + k0 + xc]);
#pragma unroll
  for (int i = 0; i < 2; ++i) {                  // 512 float4 slots each
    const int idx = tid + i * 256;
    const int wr = idx >> 2, wc = (idx & 3) * 4;
    async_ld_b128(&wqb[wr * LDP + wc], &Wq[(size_t)(n0b + wr) * DIM + k0 + wc]);
    async_ld_b128(&wkb[wr * LDP + wc], &Wk[(size_t)(n0b + wr) * DIM + k0 + wc]);
  }
}

__global__ __launch_bounds__(256) void qkproj_kernel(
    const float* __restrict__ X,
    const float* __restrict__ Wq, const float* __restrict__ bq,
    const float* __restrict__ Wk, const float* __restrict__ bk,
    float* __restrict__ Qo, float* __restrict__ Ko) {
  __shared__ float Xs [2 * 64  * LDP];
  __shared__ float Wqs[2 * 128 * LDP];
  __shared__ float Wks[2 * 128 * LDP];

  const int tid  = threadIdx.x;
  const int wid  = tid >> 5;
  const int lane = tid & 31;
  const int r    = lane & 15;
  const int h    = lane >> 4;        // half-wave selects K pair {0,1} vs {2,3}
  const int m0b  = blockIdx.x * 64;
  const int n0b  = blockIdx.y * 128;
  const int wm   = (wid & 1) * 32;   // wave M offset (2 waves x 32 rows)
  const int wn   = (wid >> 1) * 32;  // wave N offset (4 waves x 32 cols)

  const v8f vzero = {0.f, 0.f, 0.f, 0.f, 0.f, 0.f, 0.f, 0.f};
  v8f accq[2][2], acck[2][2];
#pragma unroll
  for (int i = 0; i < 2; ++i)
#pragma unroll
    for (int j = 0; j < 2; ++j) { accq[i][j] = vzero; acck[i][j] = vzero; }

  // Prologue: fill buffer 0.
  k1_issue(X, Wq, Wk, &Xs[0], &Wqs[0], &Wks[0], m0b, n0b, 0, tid);

  for (int k0 = 0; k0 < DIM; k0 += 16) {
    const int p = (k0 >> 4) & 1;
    wait_async0();
    __syncthreads();
    if (k0 + 16 < DIM)
      k1_issue(X, Wq, Wk, &Xs[(1 - p) * 64 * LDP], &Wqs[(1 - p) * 128 * LDP],
               &Wks[(1 - p) * 128 * LDP], m0b, n0b, k0 + 16, tid);

    const float* xb  = &Xs [p * 64  * LDP];
    const float* wqb = &Wqs[p * 128 * LDP];
    const float* wkb = &Wks[p * 128 * LDP];
#pragma unroll
    for (int kk = 0; kk < 4; ++kk) {
      v2f a[2], bqv[2], bkv[2];
#pragma unroll
      for (int i = 0; i < 2; ++i) {
        const int ra = (wm + i * 16 + r) * LDP + kk * 4 + 2 * h;
        a[i].x = xb[ra]; a[i].y = xb[ra + 1];
      }
#pragma unroll
      for (int j = 0; j < 2; ++j) {
        const int rw = (wn + j * 16 + r) * LDP + kk * 4 + 2 * h;
        bqv[j].x = wqb[rw]; bqv[j].y = wqb[rw + 1];
        bkv[j].x = wkb[rw]; bkv[j].y = wkb[rw + 1];
      }
#pragma unroll
      for (int i = 0; i < 2; ++i)
#pragma unroll
        for (int j = 0; j < 2; ++j) {
          accq[i][j] = wmma_f32(a[i], bqv[j], accq[i][j]);
          acck[i][j] = wmma_f32(a[i], bkv[j], acck[i][j]);
        }
    }
  }

  // Epilogue: C/D layout -> row = jj + 8*h, col = lane&15 (within 16x16 tile).
#pragma unroll
  for (int j = 0; j < 2; ++j) {
    const int col = n0b + wn + j * 16 + r;
    const float biasq = bq[col];
    const float biask = bk[col];
#pragma unroll
    for (int i = 0; i < 2; ++i)
#pragma unroll
      for (int jj = 0; jj < 8; ++jj) {
        const int row = m0b + wm + i * 16 + jj + 8 * h;
        Qo[(size_t)row * DIM + col] = accq[i][j][jj] + biasq;
        Ko[(size_t)row * DIM + col] = acck[i][j][jj] + biask;
      }
  }
}

// ---------------------------------------------------------------------------
// Kernel 2: flash attention, fp32. Block = 16 query rows, 256 threads.
// Keys in chunks of 128 (wave w owns score cols w*16..); output 16x1024 in
// registers (wave w owns cols w*128.., eight 16x16 f32 accs).
// Dynamic LDS (floats): Vs[2][8][1024] | Ks[2][128][20] | Qs[2][16][20]
//   | Ss[16][132] | Ps[16][132] | mrow[16] lrow[16] frow[16] | red[128]
// ---------------------------------------------------------------------------
__device__ __forceinline__ void qk_issue(
    const float* Qb, const float* Kb, float* qbuf, float* kbuf,
    int q0, int kc0, int d0, int tid) {
  { const int rr = tid >> 4, cc = tid & 15;      // Qs: 256 floats
    async_ld_b32(&qbuf[rr * LDP + cc], &Qb[(size_t)(q0 + rr) * DIM + d0 + cc]); }
#pragma unroll
  for (int i = 0; i < 2; ++i) {                  // Ks: 512 float4 slots
    const int idx = tid + i * 256;
    const int rr = idx >> 2, cc = (idx & 3) * 4;
    async_ld_b128(&kbuf[rr * LDP + cc], &Kb[(size_t)(kc0 + rr) * DIM + d0 + cc]);
  }
}

__device__ __forceinline__ void v_issue(
    const float* Vb, float* vbuf, int krow0, int tid) {
#pragma unroll
  for (int i = 0; i < 8; ++i) {                  // 8 rows x 1024: 2048 float4
    const int idx = tid + i * 256;
    const int rr = idx >> 8;
    const int cc = (idx & 255) * 4;
    async_ld_b128(&vbuf[rr * 1024 + cc], &Vb[(size_t)(krow0 + rr) * DIM + cc]);
  }
}

__global__ __launch_bounds__(256) void attn_kernel(
    const float* __restrict__ Q, const float* __restrict__ K,
    const float* __restrict__ V, float* __restrict__ Out) {
  extern __shared__ float smem[];
  float* Vs   = smem;                    // 2 * 8 * 1024
  float* Ks   = Vs + 2 * 8 * 1024;       // 2 * 128 * 20
  float* Qs   = Ks + 2 * 128 * LDP;      // 2 * 16 * 20
  float* Ss   = Qs + 2 * 16 * LDP;       // 16 * 132
  float* Ps   = Ss + 16 * SPITCH;        // 16 * 132
  float* mrow = Ps + 16 * SPITCH;        // 16
  float* lrow = mrow + 16;               // 16
  float* frow = lrow + 16;               // 16
  float* red  = frow + 16;               // 128 (softmax reduction scratch)

  const int tid  = threadIdx.x;
  const int wid  = tid >> 5;
  const int lane = tid & 31;
  const int r    = lane & 15;
  const int h    = lane >> 4;
  const int b    = blockIdx.y;
  const int q0   = blockIdx.x * 16;

  const float* Qb = Q + (size_t)b * SEQ * DIM;
  const float* Kb = K + (size_t)b * SEQ * DIM;
  const float* Vb = V + (size_t)b * SEQ * DIM;

  const v8f vzero = {0.f, 0.f, 0.f, 0.f, 0.f, 0.f, 0.f, 0.f};
  v8f acc[8] = {vzero, vzero, vzero, vzero, vzero, vzero, vzero, vzero};

  if (tid < 16) { mrow[tid] = -3.0e38f; lrow[tid] = 0.0f; }
  __syncthreads();

  for (int kc0 = 0; kc0 < SEQ; kc0 += 128) {
    // ---- scores S[16 x 128] = Q_tile @ K_chunk^T, pipelined over D ----
    qk_issue(Qb, Kb, &Qs[0], &Ks[0], q0, kc0, 0, tid);
    v8f sacc = vzero;
    for (int d0 = 0; d0 < DIM; d0 += 16) {
      const int p = (d0 >> 4) & 1;
      wait_async0();
      __syncthreads();
      if (d0 + 16 < DIM)
        qk_issue(Qb, Kb, &Qs[(1 - p) * 16 * LDP], &Ks[(1 - p) * 128 * LDP],
                 q0, kc0, d0 + 16, tid);
      else
        v_issue(Vb, &Vs[0], kc0, tid);  // prefetch V slab 0 across softmax

      const float* qbuf = &Qs[p * 16 * LDP];
      const float* kbuf = &Ks[p * 128 * LDP];
      v2f afr[4], bfr[4];
#pragma unroll
      for (int kk = 0; kk < 4; ++kk) {
        const int ra = r * LDP + kk * 4 + 2 * h;
        const int rb = (wid * 16 + r) * LDP + kk * 4 + 2 * h;
        afr[kk].x = qbuf[ra]; afr[kk].y = qbuf[ra + 1];
        bfr[kk].x = kbuf[rb]; bfr[kk].y = kbuf[rb + 1];
      }
#pragma unroll
      for (int kk = 0; kk < 4; ++kk) sacc = wmma_f32(afr[kk], bfr[kk], sacc);
    }
    __syncthreads();
#pragma unroll
    for (int j = 0; j < 8; ++j)
      Ss[(j + 8 * h) * SPITCH + wid * 16 + r] = sacc[j];
    __syncthreads();

    // ---- online softmax (parallel: 8 threads per query row) ----
    if (tid < 128) {
      const int row = tid >> 3, sub = tid & 7;
      float mx = -3.0e38f;
#pragma unroll
      for (int c = sub * 16; c < sub * 16 + 16; ++c)
        mx = fmaxf(mx, Ss[row * SPITCH + c]);
      red[row * 8 + sub] = mx;
    }
    __syncthreads();
    if (tid < 16) {
      const float mprev = mrow[tid];
      float mx = mprev;
#pragma unroll
      for (int s = 0; s < 8; ++s) mx = fmaxf(mx, red[tid * 8 + s]);
      mrow[tid] = mx;
      frow[tid] = __expf(mprev - mx);
    }
    __syncthreads();
    if (tid < 128) {
      const int row = tid >> 3, sub = tid & 7;
      const float mx = mrow[row];
      float s = 0.0f;
#pragma unroll
      for (int c = sub * 16; c < sub * 16 + 16; ++c) {
        const float p = __expf(Ss[row * SPITCH + c] - mx);
        Ps[row * SPITCH + c] = p;
        s += p;
      }
      red[row * 8 + sub] = s;
    }
    __syncthreads();
    if (tid < 16) {
      float l = lrow[tid] * frow[tid];
#pragma unroll
      for (int s = 0; s < 8; ++s) l += red[tid * 8 + s];
      lrow[tid] = l;
    }
    __syncthreads();

    // ---- rescale running output accumulators ----
#pragma unroll
    for (int j = 0; j < 8; ++j) {
      const float f = frow[j + 8 * h];
#pragma unroll
      for (int t = 0; t < 8; ++t) acc[t][j] *= f;
    }

    // ---- out += P[16x128] @ V_chunk[128x1024], 8-key V slabs, pipelined ----
    for (int ksl = 0; ksl < 128; ksl += 8) {
      const int p = (ksl >> 3) & 1;
      wait_async0();
      __syncthreads();
      if (ksl + 8 < 128)
        v_issue(Vb, &Vs[(1 - p) * 8 * 1024], kc0 + ksl + 8, tid);

      const float* vb = &Vs[p * 8 * 1024];
#pragma unroll
      for (int kk = 0; kk < 2; ++kk) {
        v2f a;
        a.x = Ps[r * SPITCH + ksl + kk * 4 + 2 * h + 0];
        a.y = Ps[r * SPITCH + ksl + kk * 4 + 2 * h + 1];
        v2f bfr[8];
#pragma unroll
        for (int t = 0; t < 8; ++t) {
          const int col = wid * 128 + t * 16 + r;
          bfr[t].x = vb[(kk * 4 + 2 * h + 0) * 1024 + col];
          bfr[t].y = vb[(kk * 4 + 2 * h + 1) * 1024 + col];
        }
#pragma unroll
        for (int t = 0; t < 8; ++t) acc[t] = wmma_f32(a, bfr[t], acc[t]);
      }
    }
    __syncthreads();  // PV reads done before next chunk touches Ps/Ss
  }

  // ---- epilogue: divide by row sum, write [B,S,1,D] ----
#pragma unroll
  for (int j = 0; j < 8; ++j) {
    const float inv = 1.0f / lrow[j + 8 * h];
    const int row = q0 + j + 8 * h;
#pragma unroll
    for (int t = 0; t < 8; ++t) {
      const int col = wid * 128 + t * 16 + r;
      Out[((size_t)b * SEQ + row) * DIM + col] = acc[t][j] * inv;
    }
  }
}

// ---------------------------------------------------------------------------
extern "C" void kernel_launch(void* const* d_in, const int* in_sizes, int n_in,
                              void* d_out, int out_size, void* d_ws, size_t ws_size,
                              hipStream_t stream) {
  const float* hidden = (const float*)d_in[0];
  const float* value  = (const float*)d_in[1];
  const float* Wq     = (const float*)d_in[2];
  const float* bq     = (const float*)d_in[3];
  const float* Wk     = (const float*)d_in[4];
  const float* bk     = (const float*)d_in[5];

  float* q_ws = (float*)d_ws;                      // 8192*1024 fp32 (32MB)
  float* k_ws = q_ws + (size_t)BSM * DIM;          // 8192*1024 fp32 (32MB)

  // Kernel 1: fused Q/K projection. Grid: (8192/64) x (1024/128).
  qkproj_kernel<<<dim3(BSM / 64, DIM / 128), 256, 0, stream>>>(
      hidden, Wq, bq, Wk, bk, q_ws, k_ws);

  // Kernel 2: flash attention. Grid: (4096/16) x B. Dynamic LDS ~104KB.
  const size_t smem_bytes =
      (2 * 8 * 1024 + 2 * 128 * LDP + 2 * 16 * LDP + 2 * 16 * SPITCH + 48 + 128) *
      sizeof(float);
  attn_kernel<<<dim3(SEQ / 16, BATCH), 256, smem_bytes, stream>>>(
      q_ws, k_ws, value, (float*)d_out);
}